// FNO_51075751084690
// MI455X (gfx1250) — compile-verified
//
#include <hip/hip_runtime.h>
#include <math.h>
#include <stdint.h>

typedef __attribute__((ext_vector_type(2))) float v2f;
typedef __attribute__((ext_vector_type(8))) float v8f;

constexpr int LEN  = 8192;   // sequence length
constexpr int CH   = 64;     // width W
constexpr int NB   = 32;     // batch
constexpr int PR   = NB*CH;  // 2048 (b,ch) rows
constexpr int NF   = 128;    // 2*M frequency columns (cos | sin)
constexpr int KSEG = 8;      // split-K segments for forward DFT
constexpr int KCH  = 32;     // K chunk for double-buffered forward DFT

__device__ __forceinline__ v8f wmma4(v2f a, v2f b, v8f c) {
  // V_WMMA_F32_16X16X4_F32 : full-fp32 matrix op, wave32
  return __builtin_amdgcn_wmma_f32_16x16x4_f32(false, a, false, b, (short)0, c,
                                               false, false);
}

__device__ __forceinline__ float gelu_exact(float x) {
  return 0.5f * x * (1.0f + erff(x * 0.7071067811865476f));
}

// issue one async 16-byte global->LDS copy (ASYNCcnt-tracked)
__device__ __forceinline__ void async_g2l_b128(uint32_t lds_off, uint32_t goff,
                                               const float* base) {
  asm volatile("global_load_async_to_lds_b128 %0, %1, %2"
               :: "v"(lds_off), "v"(goff), "s"(base) : "memory");
}
__device__ __forceinline__ void wait_async0() {
  asm volatile("s_wait_asynccnt 0x0" ::: "memory");
}

// ---------------- DFT basis tables -----------------------------------------
// T [n][k2]  (k2<64: cos(2*pi*n*k2/L), k2>=64: sin(2*pi*n*(k2-64)/L))
// Tt[k2][n]  transposed copy (K-contiguous for the forward GEMM)
__global__ void k_basis(float* __restrict__ T, float* __restrict__ Tt) {
  int t  = blockIdx.x * 256 + threadIdx.x;     // 8192*128 threads exactly
  int n  = t >> 7;
  int k2 = t & 127;
  int k  = k2 & 63;
  int phase = (n * k) & (LEN - 1);             // exact integer phase
  float ang = (float)phase * (6.283185307179586f / (float)LEN);
  float v = (k2 < 64) ? cosf(ang) : sinf(ang);
  T[(size_t)n * NF + k2]   = v;
  Tt[(size_t)k2 * LEN + n] = v;
}

// ---------------- encoder: h[b,w,l] = ew[w,0]*x0 + ew[w,1]*x1 + eb[w] ------
__global__ void k_encoder(const float* __restrict__ x, const float* __restrict__ ew,
                          const float* __restrict__ eb, float* __restrict__ h) {
  int t  = blockIdx.x * 256 + threadIdx.x;     // PR * (LEN/4) threads
  int p  = t >> 11;                            // row (b*64+w)
  int l4 = (t & 2047) << 2;
  int b = p >> 6, w = p & 63;
  const float4 a0 = *(const float4*)(x + ((size_t)(b*2+0))*LEN + l4);
  const float4 a1 = *(const float4*)(x + ((size_t)(b*2+1))*LEN + l4);
  float w0 = ew[w*2+0], w1 = ew[w*2+1], bias = eb[w];
  float4 o;
  o.x = w0*a0.x + w1*a1.x + bias;
  o.y = w0*a0.y + w1*a1.y + bias;
  o.z = w0*a0.z + w1*a1.z + bias;
  o.w = w0*a0.w + w1*a1.w + bias;
  *(float4*)(h + (size_t)p*LEN + l4) = o;
}

// ---------------- instance-norm stats: mean + rsqrt(var+eps) per row -------
__global__ void k_stats(const float* __restrict__ X, float* __restrict__ st) {
  int r = blockIdx.x;
  const float* row = X + (size_t)r * LEN;
  float s = 0.f, ss = 0.f;
  for (int i = threadIdx.x; i < LEN; i += 256) {
    float v = row[i]; s += v; ss += v * v;
  }
  for (int off = 16; off; off >>= 1) {
    s  += __shfl_down(s,  off, 32);
    ss += __shfl_down(ss, off, 32);
  }
  __shared__ float ps[8], pss[8];
  int wv = threadIdx.x >> 5, ln = threadIdx.x & 31;
  if (ln == 0) { ps[wv] = s; pss[wv] = ss; }
  __syncthreads();
  if (threadIdx.x == 0) {
    float S = 0.f, SS = 0.f;
    for (int i = 0; i < 8; ++i) { S += ps[i]; SS += pss[i]; }
    float mean = S * (1.0f / LEN);
    float var  = SS * (1.0f / LEN) - mean * mean;   // biased, matches jnp.var
    st[r*2+0] = mean;
    st[r*2+1] = rsqrtf(var + 1e-5f);
  }
}

// ---------------- forward partial DFT, split-K, double-buffered async ------
// Xfp[seg][p][k2] = sum_{n in seg} (h[p,n]-m)*rs * basis[n][k2]
// block: 256 thr (8 waves), covers 32 p-rows x 128 modes; wave = one N-tile,
// two p-tiles sharing each B fragment. B table staged via async-to-LDS.
__global__ void k_fwd_dft(const float* __restrict__ h, const float* __restrict__ st,
                          const float* __restrict__ Tt, float* __restrict__ Xfp) {
  __shared__ float Al[2][32][36];
  __shared__ float Bl[2][128][36];
  int p0   = blockIdx.x * 32;
  int kbeg = blockIdx.y * (LEN / KSEG);
  int tid  = threadIdx.x;
  int wave = tid >> 5, lane = tid & 31;
  int ln = lane & 15, half = lane >> 4;

  auto fill = [&](int buf, int k0) {
    // B: async copy Tt[r][k0+c] -> Bl[buf][r][c], r=0..127, c=0..31 (b128)
    uint32_t ldsB = (uint32_t)(uintptr_t)&Bl[buf][0][0];
#pragma unroll
    for (int i = 0; i < 4; ++i) {
      int idx = i * 256 + tid;            // 0..1023
      int r   = idx >> 3;                 // 0..127
      int c4  = (idx & 7) << 2;           // 0,4,...,28
      uint32_t goff = (uint32_t)(((r * LEN) + k0 + c4) * 4);
      uint32_t loff = ldsB + (uint32_t)((r * 36 + c4) * 4);
      async_g2l_b128(loff, goff, Tt);
    }
    // A: direct fill with on-the-fly instance-norm
#pragma unroll
    for (int i = 0; i < 4; ++i) {
      int idx = i * 256 + tid;            // 0..1023
      int r   = idx >> 5;                 // 0..31
      int c   = idx & 31;
      float m = st[(p0 + r) * 2], rs = st[(p0 + r) * 2 + 1];
      Al[buf][r][c] = (h[(size_t)(p0 + r) * LEN + k0 + c] - m) * rs;
    }
  };

  v8f acc0 = {0.f,0.f,0.f,0.f,0.f,0.f,0.f,0.f};
  v8f acc1 = {0.f,0.f,0.f,0.f,0.f,0.f,0.f,0.f};
  constexpr int NCHUNK = (LEN / KSEG) / KCH;     // 32
  fill(0, kbeg);
  for (int ci = 0; ci < NCHUNK; ++ci) {
    int buf = ci & 1;
    wait_async0();                // our async B copies for this chunk are in LDS
    __syncthreads();              // everyone's fills done / prev compute done
    if (ci + 1 < NCHUNK) fill(buf ^ 1, kbeg + (ci + 1) * KCH);  // prefetch
#pragma unroll
    for (int kk = 0; kk < KCH; kk += 4) {
      v2f b  = *(const v2f*)&Bl[buf][wave*16 + ln][kk + 2*half];
      v2f a0 = *(const v2f*)&Al[buf][ln][kk + 2*half];
      v2f a1 = *(const v2f*)&Al[buf][16 + ln][kk + 2*half];
      acc0 = wmma4(a0, b, acc0);
      acc1 = wmma4(a1, b, acc1);
    }
    __syncthreads();              // done reading buf before it is refilled
  }
  float* dst = Xfp + (size_t)blockIdx.y * (PR * NF);
#pragma unroll
  for (int j = 0; j < 8; ++j) {
    dst[(size_t)(p0 +      j + 8*half)*NF + wave*16 + ln] = acc0[j];
    dst[(size_t)(p0 + 16 + j + 8*half)*NF + wave*16 + ln] = acc1[j];
  }
}

__global__ void k_reduce8(const float* __restrict__ Xfp, float* __restrict__ Xf) {
  int t = blockIdx.x * 256 + threadIdx.x;      // PR*NF threads
  float s = 0.f;
#pragma unroll
  for (int i = 0; i < KSEG; ++i) s += Xfp[(size_t)i * (PR * NF) + t];
  Xf[t] = s;
}

// ---------------- complex mode mix + irfft scaling folded into Z -----------
__global__ void k_modemix(const float* __restrict__ Xf, const float* __restrict__ wr,
                          const float* __restrict__ wi, float* __restrict__ Z) {
  int b = blockIdx.x >> 6, o = blockIdx.x & 63;
  int k = threadIdx.x;                          // 64 threads: coalesced over k
  float yr = 0.f, yi = 0.f;
  for (int i = 0; i < 64; ++i) {
    float xr =  Xf[(b*64 + i)*NF + k];
    float xi = -Xf[(b*64 + i)*NF + 64 + k];     // rfft sign
    float a = wr[(i*64 + o)*64 + k];
    float c = wi[(i*64 + o)*64 + k];
    yr += xr*a - xi*c;
    yi += xr*c + xi*a;
  }
  float ck = (k == 0) ? (1.0f/LEN) : (2.0f/LEN);
  int p = b*64 + o;
  Z[p*NF + k]      = ck * yr;
  Z[p*NF + 64 + k] = (k == 0) ? 0.0f : (-(2.0f/LEN)) * yi;  // DC imag dropped
}

// ---------------- inverse partial DFT: x1[p,n] = sum_k2 Z[p,k2]*T[n,k2] ----
// block: 128 thr (4 waves), covers 32 p-rows x 64 n; wave = one N-tile,
// two p-tiles sharing each B fragment; T staged via async-to-LDS.
__global__ void k_inv_dft(const float* __restrict__ Z, const float* __restrict__ T,
                          float* __restrict__ x1) {
  __shared__ float Zl[32][132];
  __shared__ float Tl[64][132];
  int p0 = blockIdx.x * 32;
  int n0 = blockIdx.y * 64;
  int tid = threadIdx.x;
  int wave = tid >> 5, lane = tid & 31;
  int ln = lane & 15, half = lane >> 4;
  // async stage T[n0+r][c] -> Tl[r][c], 64x128 floats as b128
  {
    uint32_t ldsT = (uint32_t)(uintptr_t)&Tl[0][0];
#pragma unroll
    for (int i = 0; i < 16; ++i) {
      int idx = i * 128 + tid;           // 0..2047
      int r   = idx >> 5;                // 0..63
      int c4  = (idx & 31) << 2;         // 0..124
      uint32_t goff = (uint32_t)((((n0 + r) * NF) + c4) * 4);
      uint32_t loff = ldsT + (uint32_t)((r * 132 + c4) * 4);
      async_g2l_b128(loff, goff, T);
    }
  }
  for (int t = tid; t < 32*128; t += 128) {
    int r = t >> 7, c = t & 127;
    Zl[r][c] = Z[(p0+r)*NF + c];
  }
  wait_async0();
  __syncthreads();
  v8f acc0 = {0.f,0.f,0.f,0.f,0.f,0.f,0.f,0.f};
  v8f acc1 = {0.f,0.f,0.f,0.f,0.f,0.f,0.f,0.f};
#pragma unroll
  for (int kk = 0; kk < 128; kk += 4) {
    v2f b  = *(const v2f*)&Tl[wave*16 + ln][kk + 2*half];
    v2f a0 = *(const v2f*)&Zl[ln][kk + 2*half];
    v2f a1 = *(const v2f*)&Zl[16 + ln][kk + 2*half];
    acc0 = wmma4(a0, b, acc0);
    acc1 = wmma4(a1, b, acc1);
  }
#pragma unroll
  for (int j = 0; j < 8; ++j) {
    x1[(size_t)(p0 +      j + 8*half)*LEN + n0 + wave*16 + ln] = acc0[j];
    x1[(size_t)(p0 + 16 + j + 8*half)*LEN + n0 + wave*16 + ln] = acc1[j];
  }
}

// ---------------- fused: norm(x1) -> W1/GELU -> W2  +  Wres*h -> GELU -> h -
__global__ void k_mlp(const float* __restrict__ x1, const float* __restrict__ st,
                      float* __restrict__ h,
                      const float* __restrict__ w1, const float* __restrict__ b1,
                      const float* __restrict__ w2, const float* __restrict__ b2,
                      const float* __restrict__ ww, const float* __restrict__ wb) {
  __shared__ float Xn[64][68];   // [l][ch] normalized x1
  __shared__ float Ht[64][68];   // [l][ch] residual input h
  __shared__ float Tm[64][68];   // [l][ch] gelu(W1 xn + b1)
  int b = blockIdx.x, l0 = blockIdx.y * 64;
  int wave = threadIdx.x >> 5, lane = threadIdx.x & 31;
  int ln = lane & 15, half = lane >> 4;
  for (int t = threadIdx.x; t < 64*64; t += 256) {
    int l = t & 63, c = t >> 6;
    size_t gi = (size_t)(b*64 + c)*LEN + l0 + l;
    float m = st[(b*64+c)*2], rs = st[(b*64+c)*2+1];
    Xn[l][c] = (x1[gi] - m) * rs;
    Ht[l][c] = h[gi];
  }
  __syncthreads();
  // stage 1: Tm = gelu(W1 * Xn + b1)
  for (int pass = 0; pass < 2; ++pass) {
    int t_ = wave + 8*pass;
    int mo = (t_ >> 2) * 16, nO = (t_ & 3) * 16;
    v8f acc = {0.f,0.f,0.f,0.f,0.f,0.f,0.f,0.f};
#pragma unroll
    for (int kk = 0; kk < 64; kk += 4) {
      v2f a = *(const v2f*)&w1[(mo+ln)*64 + kk + 2*half];
      v2f bfr = *(const v2f*)&Xn[nO+ln][kk + 2*half];
      acc = wmma4(a, bfr, acc);
    }
#pragma unroll
    for (int j = 0; j < 8; ++j) {
      int m = mo + j + 8*half;
      Tm[nO+ln][m] = gelu_exact(acc[j] + b1[m]);
    }
  }
  __syncthreads();
  // stage 2: h = gelu(W2*Tm + b2 + Wres*Ht + wb)
  for (int pass = 0; pass < 2; ++pass) {
    int t_ = wave + 8*pass;
    int mo = (t_ >> 2) * 16, nO = (t_ & 3) * 16;
    v8f u = {0.f,0.f,0.f,0.f,0.f,0.f,0.f,0.f};
    v8f r = {0.f,0.f,0.f,0.f,0.f,0.f,0.f,0.f};
#pragma unroll
    for (int kk = 0; kk < 64; kk += 4) {
      v2f a2 = *(const v2f*)&w2[(mo+ln)*64 + kk + 2*half];
      v2f bt = *(const v2f*)&Tm[nO+ln][kk + 2*half];
      u = wmma4(a2, bt, u);
      v2f ar = *(const v2f*)&ww[(mo+ln)*64 + kk + 2*half];
      v2f bh = *(const v2f*)&Ht[nO+ln][kk + 2*half];
      r = wmma4(ar, bh, r);
    }
#pragma unroll
    for (int j = 0; j < 8; ++j) {
      int m = mo + j + 8*half;
      float v = u[j] + b2[m] + r[j] + wb[m];
      h[(size_t)(b*64 + m)*LEN + l0 + nO + ln] = gelu_exact(v);
    }
  }
}

// ---------------- decoder: gelu(Wd1*h + bd1) -> Wd2 dot + bd2 --------------
__global__ void k_decoder(const float* __restrict__ h,
                          const float* __restrict__ dw1, const float* __restrict__ db1,
                          const float* __restrict__ dw2, const float* __restrict__ db2,
                          float* __restrict__ out) {
  __shared__ float Ht[64][68];    // [l][ch64]
  __shared__ float D1[64][132];   // [l][ch128]
  int b = blockIdx.x, l0 = blockIdx.y * 64;
  int wave = threadIdx.x >> 5, lane = threadIdx.x & 31;
  int ln = lane & 15, half = lane >> 4;
  for (int t = threadIdx.x; t < 64*64; t += 256) {
    int l = t & 63, c = t >> 6;
    Ht[l][c] = h[(size_t)(b*64 + c)*LEN + l0 + l];
  }
  __syncthreads();
  for (int t_ = wave; t_ < 32; t_ += 8) {
    int mo = (t_ >> 2) * 16, nO = (t_ & 3) * 16;
    v8f acc = {0.f,0.f,0.f,0.f,0.f,0.f,0.f,0.f};
#pragma unroll
    for (int kk = 0; kk < 64; kk += 4) {
      v2f a = *(const v2f*)&dw1[(mo+ln)*64 + kk + 2*half];
      v2f bfr = *(const v2f*)&Ht[nO+ln][kk + 2*half];
      acc = wmma4(a, bfr, acc);
    }
#pragma unroll
    for (int j = 0; j < 8; ++j) {
      int m = mo + j + 8*half;
      D1[nO+ln][m] = gelu_exact(acc[j] + db1[m]);
    }
  }
  __syncthreads();
  if (threadIdx.x < 64) {
    int l = threadIdx.x;
    float s = db2[0];
    for (int c = 0; c < 128; ++c) s += dw2[c] * D1[l][c];
    out[(size_t)b*LEN + l0 + l] = s;
  }
}

extern "C" void kernel_launch(void* const* d_in, const int* in_sizes, int n_in,
                              void* d_out, int out_size, void* d_ws, size_t ws_size,
                              hipStream_t stream) {
  (void)in_sizes; (void)n_in; (void)out_size; (void)ws_size;
  float* ws  = (float*)d_ws;
  float* h   = ws;
  float* x1  = h   + (size_t)PR * LEN;          // 64 MB each
  float* T   = x1  + (size_t)PR * LEN;          // [LEN][NF]
  float* Tt  = T   + (size_t)LEN * NF;          // [NF][LEN]
  float* Xf  = Tt  + (size_t)LEN * NF;          // [PR][NF]
  float* Z   = Xf  + (size_t)PR * NF;           // [PR][NF]
  float* st1 = Z   + (size_t)PR * NF;           // [PR][2]
  float* st2 = st1 + 2*PR;
  float* Xfp = st2 + 2*PR;                      // [KSEG][PR][NF]

  const float* x     = (const float*)d_in[0];
  const float* enc_w = (const float*)d_in[1];
  const float* enc_b = (const float*)d_in[2];

  k_basis  <<<(LEN*NF)/256, 256, 0, stream>>>(T, Tt);
  k_encoder<<<(PR*LEN/4)/256, 256, 0, stream>>>(x, enc_w, enc_b, h);

  for (int i = 0; i < 4; ++i) {
    const float* wr  = (const float*)d_in[3 + 8*i + 0];
    const float* wi  = (const float*)d_in[3 + 8*i + 1];
    const float* ww  = (const float*)d_in[3 + 8*i + 2];
    const float* wb  = (const float*)d_in[3 + 8*i + 3];
    const float* m1  = (const float*)d_in[3 + 8*i + 4];
    const float* mb1 = (const float*)d_in[3 + 8*i + 5];
    const float* m2  = (const float*)d_in[3 + 8*i + 6];
    const float* mb2 = (const float*)d_in[3 + 8*i + 7];

    k_stats  <<<PR, 256, 0, stream>>>(h, st1);
    k_fwd_dft<<<dim3(PR/32, KSEG), 256, 0, stream>>>(h, st1, Tt, Xfp);
    k_reduce8<<<(PR*NF)/256, 256, 0, stream>>>(Xfp, Xf);
    k_modemix<<<NB*CH, 64, 0, stream>>>(Xf, wr, wi, Z);
    k_inv_dft<<<dim3(PR/32, LEN/64), 128, 0, stream>>>(Z, T, x1);
    k_stats  <<<PR, 256, 0, stream>>>(x1, st2);
    k_mlp    <<<dim3(NB, LEN/64), 256, 0, stream>>>(x1, st2, h,
                                                    m1, mb1, m2, mb2, ww, wb);
  }
  k_decoder<<<dim3(NB, LEN/64), 256, 0, stream>>>(h,
      (const float*)d_in[35], (const float*)d_in[36],
      (const float*)d_in[37], (const float*)d_in[38], (float*)d_out);
}